// kernel_generated_3_21887153341390
// MI455X (gfx1250) — compile-verified
//
#include <hip/hip_runtime.h>
#include <hip/hip_bf16.h>

// out[i,h,w] = sum_{j<3,c<64,k2<3} W[i,j,c,k2] * T[(j,c,k2),(h,w)]
// => GEMM  C[256][196] = A[256][576] * B[576][196],  A = w flat (no transform).
//
// B is stored K-pair interleaved in workspace:  Bp[k/2][n][2] = (B[k][n], B[k+1][n])
// so both the A fragment (A[m][k], A[m][k+1]) and the B fragment
// (B[k][n], B[k+1][n]) of V_WMMA_F32_16X16X4_F32 are single 8-byte loads.

#define KDIM   576
#define NVALID 196
#define NPAD   208   // 13 * 16
#define MDIM   256
#define PROW   (NPAD * 2)   // floats per K-pair row in Bp

typedef __attribute__((ext_vector_type(2))) float v2f;
typedef __attribute__((ext_vector_type(8))) float v8f;

// ---------------------------------------------------------------------------
// Stage 1: build Bp (pair-interleaved B) from x (1,128,14,14).
//   logical row = j*192 + c*3 + k2 ; col = h*14 + w
//   wq = w + k2 - 1            (second unfold, zero pad outside [0,14))
//   wp = (wq - 1) mod 14       (roll +1 along w)
//   term1 = xp(64+c, (h-1)%14 + j - 1, wp)   (zero pad in H)
//   term2 = xp(c,     h       + j - 1, wp)
//   store at Bp[(row>>1)*PROW + col*2 + (row&1)]
// ---------------------------------------------------------------------------
__global__ void build_b_kernel(const float* __restrict__ x, float* __restrict__ Bp) {
    int idx = blockIdx.x * blockDim.x + threadIdx.x;
    if (idx >= KDIM * NPAD) return;
    int row = idx / NPAD;
    int col = idx - row * NPAD;

    float v = 0.0f;
    if (col < NVALID) {
        int h  = col / 14;
        int w  = col - h * 14;
        int j  = row / 192;
        int rm = row - j * 192;
        int c  = rm / 3;
        int k2 = rm - c * 3;

        int wq = w + k2 - 1;                 // zero pad outside [0,14)
        if (wq >= 0 && wq < 14) {
            int wp = wq - 1; if (wp < 0) wp += 14;   // cyclic roll by +1 (w)
            int h1 = h - 1;  if (h1 < 0) h1 += 14;   // cyclic (h,c_hi) roll
            int hh1 = h1 + j - 1;                    // first unfold (zero pad)
            int hh2 = h  + j - 1;
            float t1 = (hh1 >= 0 && hh1 < 14) ? x[(64 + c) * 196 + hh1 * 14 + wp] : 0.0f;
            float t2 = (hh2 >= 0 && hh2 < 14) ? x[c * 196 + hh2 * 14 + wp] : 0.0f;
            v = t1 + t2;
        }
    }
    Bp[(row >> 1) * PROW + col * 2 + (row & 1)] = v;
}

// ---------------------------------------------------------------------------
// Stage 2: WMMA f32 GEMM. One wave (32 threads) per 16x16 output tile.
// grid = (M/16, NPAD/16) = (16, 13). K loop: 144 x V_WMMA_F32_16X16X4_F32.
// Full f32 precision (matches the f32 reference; no bf16/f16 downcast error).
// ---------------------------------------------------------------------------
__global__ void __launch_bounds__(32)
wmma_gemm_f32_kernel(const float* __restrict__ A,    // [256][576]  (= w, flat)
                     const float* __restrict__ Bp,   // [288][208][2] pair-interleaved
                     float* __restrict__ C) {        // [256][196]
    const int lane = threadIdx.x & 31;
    const int half = lane >> 4;      // 0: K pair {k,k+1}, 1: {k+2,k+3}
    const int l    = lane & 15;

    const int tileM = blockIdx.x;    // 0..15
    const int tileN = blockIdx.y;    // 0..12

    const int arow = tileM * 16 + l;         // A fragment: M = lane&15
    const int bcol = tileN * 16 + l;         // B fragment: N = lane&15

    const float* __restrict__ Arow  = A + arow * KDIM + half * 2;
    const float* __restrict__ Bbase = Bp + (size_t)half * PROW + bcol * 2;

    v8f acc = {};
#pragma unroll 4
    for (int k = 0; k < KDIM; k += 4) {
        // A fragment: (A[m][kb], A[m][kb+1]) — one b64 load per lane
        v2f a = *(const v2f*)(Arow + k);
        // B fragment: (B[kb][n], B[kb+1][n]) — one b64 load per lane
        v2f b = *(const v2f*)(Bbase + (size_t)(k >> 1) * PROW);
        // D = A(16x4,f32) * B(4x16,f32) + C(16x16,f32)
        acc = __builtin_amdgcn_wmma_f32_16x16x4_f32(
            /*neg_a=*/false, a, /*neg_b=*/false, b,
            /*c_mod=*/(short)0, acc, /*reuse_a=*/false, /*reuse_b=*/false);
    }

    // C/D layout: acc[r] -> row = r + 8*half, col = lane&15
    const int col = tileN * 16 + l;
    if (col < NVALID) {
        const int rowBase = tileM * 16 + half * 8;
#pragma unroll
        for (int r = 0; r < 8; ++r) {
            C[(rowBase + r) * NVALID + col] = acc[r];
        }
    }
}

extern "C" void kernel_launch(void* const* d_in, const int* in_sizes, int n_in,
                              void* d_out, int out_size, void* d_ws, size_t ws_size,
                              hipStream_t stream) {
    const float* x = (const float*)d_in[0];   // (1,128,14,14)  = 25088 f32
    const float* w = (const float*)d_in[1];   // (256,3,64,3)   = 147456 f32 = A[256][576]
    float* out = (float*)d_out;               // (1,256,14,14)  = 50176 f32
    float* Bp  = (float*)d_ws;                // 576*208 f32 = 479232 bytes (pair-interleaved)

    {
        const int total   = KDIM * NPAD;      // 119808
        const int threads = 256;
        const int blocks  = (total + threads - 1) / threads;
        build_b_kernel<<<blocks, threads, 0, stream>>>(x, Bp);
    }
    {
        dim3 grid(MDIM / 16, NPAD / 16);      // (16, 13)
        wmma_gemm_f32_kernel<<<grid, 32, 0, stream>>>(w, Bp, out);
    }
}